// Supernode_43267500540335
// MI455X (gfx1250) — compile-verified
//
#include <hip/hip_runtime.h>
#include <hip/hip_bf16.h>

typedef __attribute__((ext_vector_type(2))) float v2f;
typedef __attribute__((ext_vector_type(8))) float v8f;

#define CIN 32
#define COUT 32

// ---------------------------------------------------------------------------
// Degree / normalization kernels
// ---------------------------------------------------------------------------
__global__ void deg_init_kernel(float* __restrict__ deg, int n) {
    int i = blockIdx.x * blockDim.x + threadIdx.x;
    if (i < n) deg[i] = 1.0f;  // self-loop contribution
}

__global__ void deg_accum_kernel(const int* __restrict__ dst, float* __restrict__ deg, int e) {
    int i = blockIdx.x * blockDim.x + threadIdx.x;
    if (i < e) atomicAdd(&deg[dst[i]], 1.0f);
}

__global__ void dinv_kernel(float* __restrict__ deg, int n) {
    int i = blockIdx.x * blockDim.x + threadIdx.x;
    if (i < n) deg[i] = rsqrtf(deg[i]);  // deg >= 1 always (self-loop)
}

// ---------------------------------------------------------------------------
// Stage W[32x32] into LDS pre-swizzled into B-fragment order:
// flat fragment index p = (k*2 + nh)*32 + lane, lane -> (half = lane>>4,
// l16 = lane&15); fragment = { W[4k+2*half][nh*16+l16], W[4k+2*half+1][...] }.
// Each lane then fetches its float2 with ONE contiguous ds_load_b64.
// ---------------------------------------------------------------------------
__device__ __forceinline__ void stage_w_swizzled(const float* __restrict__ W,
                                                 float* __restrict__ Wl) {
    for (int p = threadIdx.x; p < 512; p += blockDim.x) {
        const int k    = p >> 6;          // 8 K-steps
        const int rem  = p & 63;
        const int nh   = rem >> 5;        // output N-half
        const int lane = rem & 31;
        const int half = lane >> 4;
        const int l16  = lane & 15;
        const int row  = 4 * k + 2 * half;
        const int col  = nh * 16 + l16;
        Wl[2 * p + 0] = W[(row + 0) * COUT + col];
        Wl[2 * p + 1] = W[(row + 1) * COUT + col];
    }
}

// ---------------------------------------------------------------------------
// WMMA GEMM: Y[n,32] = X[n,32] @ W[32,32]   (f32, 16x16x4 WMMA)
// One wave -> 16 rows x 32 cols (two 16x16 C tiles, 8 K-steps of 4).
// ---------------------------------------------------------------------------
__global__ void gemm32_wmma_kernel(const float* __restrict__ X, const float* __restrict__ W,
                                   float* __restrict__ Y, int n) {
    __shared__ float Wl[2 * 512];
    stage_w_swizzled(W, Wl);
    __syncthreads();

    const int wave = threadIdx.x >> 5;
    const int lane = threadIdx.x & 31;
    const int row0 = (blockIdx.x * (blockDim.x >> 5) + wave) * 16;
    if (row0 >= n) return;

    const int half = lane >> 4;   // 0: lanes 0-15 (K pair 0,1) ; 1: lanes 16-31 (K pair 2,3)
    const int l16  = lane & 15;
    const bool interior = (row0 + 16 <= n);   // wave-uniform

    int r = row0 + l16; if (r >= n) r = n - 1;   // clamped A-row (store is guarded)
    const float* __restrict__ Xr = X + (size_t)r * CIN + 2 * half;
    const float* __restrict__ Wf = Wl + 2 * lane;

    v8f c0 = {}; v8f c1 = {};
#pragma unroll
    for (int k = 0; k < 8; ++k) {
        v2f a  = *(const v2f*)(Xr + 4 * k);
        v2f b0 = *(const v2f*)(Wf + (k * 2 + 0) * 64);   // ds_load_b64, imm offset
        v2f b1 = *(const v2f*)(Wf + (k * 2 + 1) * 64);
        c0 = __builtin_amdgcn_wmma_f32_16x16x4_f32(false, a, false, b0, (short)0, c0, false, false);
        c1 = __builtin_amdgcn_wmma_f32_16x16x4_f32(false, a, false, b1, (short)0, c1, false, false);
    }

    float* __restrict__ Yp = Y + (size_t)(row0 + 8 * half) * COUT + l16;
    if (interior) {                    // fast path: uniform branch, unguarded stores
#pragma unroll
        for (int v = 0; v < 8; ++v) {
            Yp[(size_t)v * COUT]      = c0[v];
            Yp[(size_t)v * COUT + 16] = c1[v];
        }
    } else {
#pragma unroll
        for (int v = 0; v < 8; ++v) {
            const int row = row0 + v + 8 * half;
            if (row < n) {
                Y[(size_t)row * COUT + l16]      = c0[v];
                Y[(size_t)row * COUT + 16 + l16] = c1[v];
            }
        }
    }
}

// ---------------------------------------------------------------------------
// acc[i,c] = h[i,c] * dinv[i]^2   (self-loop message, no atomics needed)
// float4-vectorized: one thread per 4 channels.
// ---------------------------------------------------------------------------
__global__ void selfloop_init_kernel(const float* __restrict__ h, const float* __restrict__ dinv,
                                     float* __restrict__ acc, int n) {
    int t = blockIdx.x * blockDim.x + threadIdx.x;   // over n * 8 float4 groups
    if (t < n * (CIN / 4)) {
        int i = t >> 3;
        float di = dinv[i];
        float w = di * di;
        float4 v = ((const float4*)h)[t];
        v.x *= w; v.y *= w; v.z *= w; v.w *= w;
        ((float4*)acc)[t] = v;
    }
}

// ---------------------------------------------------------------------------
// Edge scatter: one lane per (edge, channel). Lanes of a wave cover all 32
// channels of one edge -> coalesced gather + coalesced f32 atomics into L2.
// ---------------------------------------------------------------------------
__global__ void edge_scatter_kernel(const float* __restrict__ h, const float* __restrict__ dinv,
                                    const int* __restrict__ src, const int* __restrict__ dst,
                                    float* __restrict__ acc, int e) {
    long long t = (long long)blockIdx.x * blockDim.x + threadIdx.x;
    int ei = (int)(t >> 5);
    if (ei >= e) return;
    int c = (int)(t & 31);
    int s = src[ei];
    int d = dst[ei];
    float w = dinv[s] * dinv[d];
    atomicAdd(&acc[(size_t)d * CIN + c], h[(size_t)s * CIN + c] * w);
}

// ---------------------------------------------------------------------------
// h[i,c] = relu(h[i,c] + b[c])   float4-vectorized
// ---------------------------------------------------------------------------
__global__ void bias_relu_kernel(float* __restrict__ h, const float* __restrict__ b, int n4) {
    int t = blockIdx.x * blockDim.x + threadIdx.x;   // over n * 8 float4 groups
    if (t < n4) {
        float4 bv = ((const float4*)b)[t & 7];
        float4 v  = ((float4*)h)[t];
        v.x = fmaxf(v.x + bv.x, 0.0f);
        v.y = fmaxf(v.y + bv.y, 0.0f);
        v.z = fmaxf(v.z + bv.z, 0.0f);
        v.w = fmaxf(v.w + bv.w, 0.0f);
        ((float4*)h)[t] = v;
    }
}

// ---------------------------------------------------------------------------
// Fused tail: out[i,:] = out[i,:] + b2 + nf[i,:]@Wn + pf[i,:]@Wt
// Seeds the WMMA accumulators with the scattered layer-2 result + bias,
// then runs two 8-step K chains per 16x16 tile (32 WMMAs per wave).
// ---------------------------------------------------------------------------
__global__ void adapter_wmma_kernel(const float* __restrict__ nf, const float* __restrict__ pf,
                                    const float* __restrict__ Wn, const float* __restrict__ Wt,
                                    const float* __restrict__ b2, float* __restrict__ out, int n) {
    __shared__ float Wl[2 * 1024];
    stage_w_swizzled(Wn, Wl);
    stage_w_swizzled(Wt, Wl + 1024);
    __syncthreads();

    const int wave = threadIdx.x >> 5;
    const int lane = threadIdx.x & 31;
    const int row0 = (blockIdx.x * (blockDim.x >> 5) + wave) * 16;
    if (row0 >= n) return;

    const int half = lane >> 4;
    const int l16  = lane & 15;
    const bool interior = (row0 + 16 <= n);   // wave-uniform

    const float bias0 = b2[l16];
    const float bias1 = b2[16 + l16];

    float* __restrict__ Op = out + (size_t)(row0 + 8 * half) * COUT + l16;

    v8f c0, c1;
    if (interior) {                    // fast path: unguarded seed loads
#pragma unroll
        for (int v = 0; v < 8; ++v) {
            c0[v] = Op[(size_t)v * COUT]      + bias0;
            c1[v] = Op[(size_t)v * COUT + 16] + bias1;
        }
    } else {
#pragma unroll
        for (int v = 0; v < 8; ++v) {
            int row = row0 + v + 8 * half; if (row >= n) row = n - 1;
            c0[v] = out[(size_t)row * COUT + l16]      + bias0;
            c1[v] = out[(size_t)row * COUT + 16 + l16] + bias1;
        }
    }

    int r = row0 + l16; if (r >= n) r = n - 1;
    const float* __restrict__ Ar = nf + (size_t)r * CIN + 2 * half;
    const float* __restrict__ Br = pf + (size_t)r * CIN + 2 * half;
    const float* __restrict__ Wf = Wl + 2 * lane;

#pragma unroll
    for (int k = 0; k < 8; ++k) {     // += nf @ Wn
        v2f a  = *(const v2f*)(Ar + 4 * k);
        v2f b0 = *(const v2f*)(Wf + (k * 2 + 0) * 64);
        v2f b1 = *(const v2f*)(Wf + (k * 2 + 1) * 64);
        c0 = __builtin_amdgcn_wmma_f32_16x16x4_f32(false, a, false, b0, (short)0, c0, false, false);
        c1 = __builtin_amdgcn_wmma_f32_16x16x4_f32(false, a, false, b1, (short)0, c1, false, false);
    }
#pragma unroll
    for (int k = 0; k < 8; ++k) {     // += pf @ Wt
        v2f a  = *(const v2f*)(Br + 4 * k);
        v2f b0 = *(const v2f*)(Wf + 1024 + (k * 2 + 0) * 64);
        v2f b1 = *(const v2f*)(Wf + 1024 + (k * 2 + 1) * 64);
        c0 = __builtin_amdgcn_wmma_f32_16x16x4_f32(false, a, false, b0, (short)0, c0, false, false);
        c1 = __builtin_amdgcn_wmma_f32_16x16x4_f32(false, a, false, b1, (short)0, c1, false, false);
    }

    if (interior) {
#pragma unroll
        for (int v = 0; v < 8; ++v) {
            Op[(size_t)v * COUT]      = c0[v];
            Op[(size_t)v * COUT + 16] = c1[v];
        }
    } else {
#pragma unroll
        for (int v = 0; v < 8; ++v) {
            const int row = row0 + v + 8 * half;
            if (row < n) {
                out[(size_t)row * COUT + l16]      = c0[v];
                out[(size_t)row * COUT + 16 + l16] = c1[v];
            }
        }
    }
}

// ---------------------------------------------------------------------------
// Launch
// ---------------------------------------------------------------------------
extern "C" void kernel_launch(void* const* d_in, const int* in_sizes, int n_in,
                              void* d_out, int out_size, void* d_ws, size_t ws_size,
                              hipStream_t stream) {
    const int N = in_sizes[0] / CIN;
    const int E = in_sizes[1] / 2;

    const float* x   = (const float*)d_in[0];
    const int*   ei  = (const int*)d_in[1];
    const float* nf  = (const float*)d_in[2];
    const float* pf  = (const float*)d_in[3];
    const float* W1  = (const float*)d_in[4];
    const float* b1  = (const float*)d_in[5];
    const float* W2  = (const float*)d_in[6];
    const float* b2  = (const float*)d_in[7];
    const float* Wn  = (const float*)d_in[8];
    const float* Wt  = (const float*)d_in[9];
    const int* src = ei;
    const int* dst = ei + E;

    float* ws   = (float*)d_ws;
    float* dinv = ws;                                        // N floats (deg -> dinv in place)
    float* bufA = ws + (((size_t)N + 255) & ~(size_t)255);   // N*32: h0, later h2lin
    float* bufB = bufA + (size_t)N * CIN;                    // N*32: acc1 -> h1
    float* out  = (float*)d_out;

    const int T = 256;
    const dim3 blk(T);
    const dim3 grN((N + T - 1) / T);
    const dim3 grE((E + T - 1) / T);
    const dim3 grN4(((size_t)N * (CIN / 4) + T - 1) / T);    // float4-vectorized passes
    const dim3 grEC(((size_t)E * CIN + T - 1) / T);
    const dim3 grG((N + 8 * 16 - 1) / (8 * 16));             // 8 waves/block, 16 rows/wave

    // degrees + symmetric normalization
    deg_init_kernel<<<grN, blk, 0, stream>>>(dinv, N);
    deg_accum_kernel<<<grE, blk, 0, stream>>>(dst, dinv, E);
    dinv_kernel<<<grN, blk, 0, stream>>>(dinv, N);

    // layer 1: h0 = x@W1 ; acc1 = scatter(norm * h0) ; h1 = relu(acc1 + b1)
    gemm32_wmma_kernel<<<grG, blk, 0, stream>>>(x, W1, bufA, N);
    selfloop_init_kernel<<<grN4, blk, 0, stream>>>(bufA, dinv, bufB, N);
    edge_scatter_kernel<<<grEC, blk, 0, stream>>>(bufA, dinv, src, dst, bufB, E);
    bias_relu_kernel<<<grN4, blk, 0, stream>>>(bufB, b1, N * (CIN / 4));

    // layer 2: h2lin = h1@W2 ; out = scatter(norm * h2lin)
    gemm32_wmma_kernel<<<grG, blk, 0, stream>>>(bufB, W2, bufA, N);
    selfloop_init_kernel<<<grN4, blk, 0, stream>>>(bufA, dinv, out, N);
    edge_scatter_kernel<<<grEC, blk, 0, stream>>>(bufA, dinv, src, dst, out, E);

    // tail: out += b2 + nf@Wn + pf@Wt  (fused WMMA accumulation)
    adapter_wmma_kernel<<<grG, blk, 0, stream>>>(nf, pf, Wn, Wt, b2, out, N);
}